// PushPullLoss_88673894793881
// MI455X (gfx1250) — compile-verified
//
#include <hip/hip_runtime.h>

#define LVL 16
#define NPTS 9261
#define TT 30
#define KC 10
#define FF 4
#define VQ (1 << 19)
#define VC (1 << 16)
#define WPB 8                      // waves per block (256 threads)
#define PTOT (LVL * NPTS)          // 148176 = 8 * 18522
#define NBLK (PTOT / WPB)          // 18522

typedef __attribute__((ext_vector_type(2))) float v2f;
typedef __attribute__((ext_vector_type(8))) float v8f;

// group-of-32 ds_swizzle: offset = (xor_mask<<10) | 0x1F
template <int XORM>
__device__ __forceinline__ float swz_xor(float x) {
    int i = __builtin_amdgcn_ds_swizzle(__float_as_int(x), (XORM << 10) | 0x1F);
    return __int_as_float(i);
}

__device__ __forceinline__ float wave_reduce_add(float x) {
    x += swz_xor<1>(x);
    x += swz_xor<2>(x);
    x += swz_xor<4>(x);
    x += swz_xor<8>(x);
    x += swz_xor<16>(x);
    return x;
}

__device__ __forceinline__ float fast_rcp(float x) {
    return __builtin_amdgcn_rcpf(x);   // v_rcp_f32 (TRANS unit, ~1 ulp)
}

__device__ __forceinline__ v8f wmma_f32_16x16x4(v2f a, v2f b, v8f c) {
    return __builtin_amdgcn_wmma_f32_16x16x4_f32(
        /*neg_a=*/false, a, /*neg_b=*/false, b,
        /*c_mod=*/(short)0, c, /*reuse_a=*/false, /*reuse_b=*/false);
}

__global__ void __launch_bounds__(256) push_pull_kernel(
    const float* __restrict__ xyzt_tables,     // [L, 2^19, 4]
    const float* __restrict__ centroid_tables, // [K, L, 2^16, 4]
    const int*   __restrict__ xyzt_idx,        // [L, 9261, 30]
    const int*   __restrict__ xyz_idx,         // [L, 9261]
    float*       __restrict__ block_out)       // [NBLK]
{
    __shared__ float4 s_emb[WPB][32];   // emb rows, padded (rows 30,31 = 0)
    __shared__ float  s_en[WPB][32];    // ||emb||^2
    __shared__ float4 s_cent[WPB][16];  // centroid rows, padded (10..15 = 0)
    __shared__ float  s_cn[WPB][16];    // ||cent||^2
    __shared__ float4 s_sc[WPB][32];    // soft centroids (rows 30,31 = 0)
    __shared__ float  s_sn[WPB][32];    // ||sc||^2
    __shared__ float  s_partial[WPB];

    const int lane = threadIdx.x & 31;
    const int wave = threadIdx.x >> 5;
    const int p    = blockIdx.x * WPB + wave;  // always < PTOT (exact divide)
    const int level = p / NPTS;
    const int n     = p - level * NPTS;
    const int h = lane >> 4;    // lane half
    const int q = lane & 15;    // lane within half
    const float hmask = (h == 0) ? 1.f : 0.f;

    const float4 z4 = make_float4(0.f, 0.f, 0.f, 0.f);

    // ---- Phase A: gather embeddings + centroids into LDS ----
    if (lane < TT) {
        int idx = xyzt_idx[(level * NPTS + n) * TT + lane];
        const float4 e = *(const float4*)(xyzt_tables +
                          ((size_t)level * VQ + (size_t)idx) * FF);
        s_emb[wave][lane] = e;
        s_en[wave][lane]  = e.x * e.x + e.y * e.y + e.z * e.z + e.w * e.w;
    } else {
        s_emb[wave][lane] = z4;
        s_en[wave][lane]  = 0.f;
    }
    if (lane < 16) {
        if (lane < KC) {
            int ci = xyz_idx[level * NPTS + n];
            const float4 c = *(const float4*)(centroid_tables +
                              (((size_t)lane * LVL + level) * VC + (size_t)ci) * FF);
            s_cent[wave][lane] = c;
            s_cn[wave][lane]   = c.x * c.x + c.y * c.y + c.z * c.z + c.w * c.w;
        } else {
            s_cent[wave][lane] = z4;
            s_cn[wave][lane]   = 0.f;
        }
    }
    asm volatile("s_wait_dscnt 0x0" ::: "memory");  // wave-lockstep LDS visibility

    // per-lane centroid slice: registers j -> k = j + 8*h
    float  cnv[8], kmask[8];
    float4 cv[8];
#pragma unroll
    for (int j = 0; j < 8; ++j) {
        int k    = j + 8 * h;
        cnv[j]   = s_cn[wave][k];
        cv[j]    = s_cent[wave][k];
        kmask[j] = (k < KC) ? 1.f : 0.f;
    }

    // ---- Phase B: G = cents(16x4) @ emb^T(4x32)  -> 2 WMMAs ----
    const float* crow = (const float*)&s_cent[wave][q];
    v2f ac;  ac.x  = crow[2 * h];            ac.y  = crow[2 * h + 1];
    const float* er0 = (const float*)&s_emb[wave][q];
    const float* er1 = (const float*)&s_emb[wave][16 + q];
    v2f be0; be0.x = er0[2 * h];             be0.y = er0[2 * h + 1];
    v2f be1; be1.x = er1[2 * h];             be1.y = er1[2 * h + 1];

    v8f zc = {0.f, 0.f, 0.f, 0.f, 0.f, 0.f, 0.f, 0.f};
    v8f G0 = wmma_f32_16x16x4(ac, be0, zc);  // D[k][t], t = 0..15
    v8f G1 = wmma_f32_16x16x4(ac, be1, zc);  // D[k][t], t = 16..31

    float pull_acc = 0.f;
#pragma unroll
    for (int nt = 0; nt < 2; ++nt) {
        v8f G = nt ? G1 : G0;
        const int t = nt * 16 + q;
        const float tmask = (t < TT) ? 1.f : 0.f;
        const float en_t = s_en[wave][t];
        float sa[8];
        float denom = 0.f;
#pragma unroll
        for (int j = 0; j < 8; ++j) {
            float d2 = en_t + cnv[j] - 2.f * G[j];
            float e  = __expf(-d2 * 5.0e7f) * kmask[j];  // -d2 / (2*sigma^2), masked
            sa[j] = e;
            denom += e;
        }
        denom += swz_xor<16>(denom);            // combine k=0..7 with k=8..9
        float inv = fast_rcp(denom + 1e-10f);
        float4 scp = z4;
#pragma unroll
        for (int j = 0; j < 8; ++j) {
            float w = sa[j] * inv;
            scp.x += w * cv[j].x;  scp.y += w * cv[j].y;
            scp.z += w * cv[j].z;  scp.w += w * cv[j].w;
        }
        scp.x += swz_xor<16>(scp.x);  scp.y += swz_xor<16>(scp.y);
        scp.z += swz_xor<16>(scp.z);  scp.w += swz_xor<16>(scp.w);
        // zero rows t >= TT so WMMA2 operands & pull term self-mask
        scp.x *= tmask;  scp.y *= tmask;  scp.z *= tmask;  scp.w *= tmask;

        // pull term (count each t once via hmask; emb rows>=TT are zero)
        {
            float4 e = s_emb[wave][t];
            float dx = e.x - scp.x, dy = e.y - scp.y;
            float dz = e.z - scp.z, dw = e.w - scp.w;
            pull_acc += hmask * (dx * dx + dy * dy + dz * dz + dw * dw);
        }
        // both halves hold identical scp -> duplicate same-value store is fine
        s_sc[wave][t] = scp;
        s_sn[wave][t] = scp.x * scp.x + scp.y * scp.y +
                        scp.z * scp.z + scp.w * scp.w;
    }
    asm volatile("s_wait_dscnt 0x0" ::: "memory");

    // ---- Phase C: S = sc @ sc^T (32x32) -> 4 WMMAs; push term ----
    v2f op[2];
    {
        const float* r0 = (const float*)&s_sc[wave][q];
        const float* r1 = (const float*)&s_sc[wave][16 + q];
        op[0].x = r0[2 * h];  op[0].y = r0[2 * h + 1];
        op[1].x = r1[2 * h];  op[1].y = r1[2 * h + 1];
    }
    const float sn_s[2] = { s_sn[wave][q], s_sn[wave][16 + q] };
    const float smask[2] = { (q < TT) ? 1.f : 0.f, (16 + q < TT) ? 1.f : 0.f };

    float push_acc = 0.f;
#pragma unroll
    for (int mt = 0; mt < 2; ++mt) {
        float snt[8], tmsk[8];
#pragma unroll
        for (int j = 0; j < 8; ++j) {
            int t   = mt * 16 + j + 8 * h;
            snt[j]  = s_sn[wave][t];
            tmsk[j] = (t < TT) ? 1.f : 0.f;
        }
#pragma unroll
        for (int nt = 0; nt < 2; ++nt) {
            v8f S = wmma_f32_16x16x4(op[mt], op[nt], zc);
#pragma unroll
            for (int j = 0; j < 8; ++j) {
                float dd = snt[j] + sn_s[nt] - 2.f * S[j];
                dd = fmaxf(dd, 0.f);
                float m = smask[nt] * tmsk[j];
                push_acc += m * fast_rcp(1e-15f + 0.5f * dd);
            }
        }
    }

    // ---- Phase D: reductions (deterministic, no atomics) ----
    const float PULL_NORM = (float)(1.0 / (9261.0 * 30.0));
    const float PUSH_NORM = (float)(1.0 / (9261.0 * 30.0 * 30.0));
    float wave_sum = pull_acc * PULL_NORM + push_acc * PUSH_NORM;
    wave_sum = wave_reduce_add(wave_sum);
    if (lane == 0) s_partial[wave] = wave_sum;
    __syncthreads();
    if (threadIdx.x == 0) {
        float acc = 0.f;
#pragma unroll
        for (int w = 0; w < WPB; ++w) acc += s_partial[w];
        block_out[blockIdx.x] = acc;
    }
}

__global__ void __launch_bounds__(256) reduce_kernel(
    const float* __restrict__ in, int n, float* __restrict__ out)
{
    __shared__ float sh[256];
    float acc = 0.f;
    for (int i = threadIdx.x; i < n; i += 256) acc += in[i];
    sh[threadIdx.x] = acc;
    __syncthreads();
    for (int s = 128; s > 0; s >>= 1) {
        if (threadIdx.x < s) sh[threadIdx.x] += sh[threadIdx.x + s];
        __syncthreads();
    }
    if (threadIdx.x == 0) out[0] = sh[0];
}

extern "C" void kernel_launch(void* const* d_in, const int* in_sizes, int n_in,
                              void* d_out, int out_size, void* d_ws, size_t ws_size,
                              hipStream_t stream) {
    const float* xyzt_tables     = (const float*)d_in[0];
    const float* centroid_tables = (const float*)d_in[1];
    const int*   xyzt_idx        = (const int*)d_in[2];
    const int*   xyz_idx         = (const int*)d_in[3];
    float* out = (float*)d_out;
    float* ws  = (float*)d_ws;   // NBLK floats of per-block partials

    push_pull_kernel<<<NBLK, 256, 0, stream>>>(
        xyzt_tables, centroid_tables, xyzt_idx, xyz_idx, ws);
    reduce_kernel<<<1, 256, 0, stream>>>(ws, NBLK, out);
}